// GCNNetwork_1082331758967
// MI455X (gfx1250) — compile-verified
//
#include <hip/hip_runtime.h>
#include <hip/hip_bf16.h>
#include <math.h>

// ---------------------------------------------------------------------------
// GCN forward for MI455X (gfx1250, wave32).
// - Dense matmuls: V_WMMA_F32_16X16X4_F32 (fp32 end-to-end, matches reference).
// - GENConv softmax aggregation: fused online-softmax over the implicit
//   complete-graph edge set, with xs/W tiles staged into LDS via the CDNA5
//   async global->LDS path (GLOBAL_LOAD_ASYNC_TO_LDS_*, ASYNCcnt waits),
//   double-buffered, 4 destination columns per block (4x less L2 traffic).
// Everything fits in L2 (192MB) -> latency/L2 bound; HBM is irrelevant here.
// ---------------------------------------------------------------------------

typedef float v2f __attribute__((ext_vector_type(2)));
typedef float v8f __attribute__((ext_vector_type(8)));

#define NODES 512
#define HID   128
#define GEN_EPS 1e-7f
#define BN_EPS  1e-5f

// ---------------- WMMA GEMM: C[M,N] = A[M,K] @ B[K,N] (row major fp32) ------
// MODE: 0 = plain, 1 = +bias, 2 = +bias +relu
// One wave (32 threads) per 16x16 output tile; K stepped by 4.
template <int MODE>
__global__ void gemm_f32_wmma(const float* __restrict__ A,
                              const float* __restrict__ B,
                              const float* __restrict__ bias,
                              float* __restrict__ C,
                              int M, int N, int K) {
  const int tilesN = N >> 4;
  const int tm = (blockIdx.x / tilesN) << 4;
  const int tn = (blockIdx.x % tilesN) << 4;
  const int lane = threadIdx.x & 31;
  const int hl   = lane >> 4;     // 0: lanes 0-15, 1: lanes 16-31
  const int l15  = lane & 15;

  v8f acc = {};
  for (int k = 0; k < K; k += 4) {
    // A 16x4: lane = M row; VGPR0/1 = K { k+2*hl, k+2*hl+1 }
    const float* Ap = A + (size_t)(tm + l15) * K + k + (hl << 1);
    v2f a; a.x = Ap[0]; a.y = Ap[1];
    // B 4x16: VGPR v holds row (k + v + 2*hl), col = l15
    const float* Bp = B + (size_t)(k + (hl << 1)) * N + tn + l15;
    v2f b; b.x = Bp[0]; b.y = Bp[N];
    acc = __builtin_amdgcn_wmma_f32_16x16x4_f32(false, a, false, b,
                                                (short)0, acc, false, false);
  }
#pragma unroll
  for (int r = 0; r < 8; ++r) {
    const int row = tm + r + (hl << 3);
    const int col = tn + l15;
    float v = acc[r];
    if (MODE >= 1) v += bias[col];
    if (MODE == 2) v = fmaxf(v, 0.0f);
    C[(size_t)row * N + col] = v;
  }
}

// ---------- CDNA5 async global -> LDS helpers -------------------------------
__device__ __forceinline__ void async_load_b128(unsigned lds_off, const float* g) {
  asm volatile("global_load_async_to_lds_b128 %0, %1, off"
               :: "v"(lds_off), "v"(g) : "memory");
}
__device__ __forceinline__ void async_load_b32(unsigned lds_off, const float* g) {
  asm volatile("global_load_async_to_lds_b32 %0, %1, off"
               :: "v"(lds_off), "v"(g) : "memory");
}

// ---------- fused GENConv message + per-dst softmax aggregation -------------
// out[j,c] = sum_i softmax_i(m) * m + xd[j,c],
// m = relu(xs[i,c] + (W[i,j]/total) * le[c]) + eps
// Block: 128 threads (thread == channel c), JT=4 destination columns.
// xs rows + W sub-columns streamed through LDS with double-buffered async
// copies; each wave issues exactly 9 async instructions per chunk, so
// s_wait_asynccnt 0x9 drains the previous chunk (in-order completion).
#define I_CHUNK 32
#define JT 4

__global__ void __launch_bounds__(128)
gen_msg_softmax(const float* __restrict__ xs,
                const float* __restrict__ xd,
                const float* __restrict__ W,     // [n,n]
                const float* __restrict__ total,
                const float* __restrict__ le,    // [HID]
                float* __restrict__ out, int n) {
  __shared__ float xs_sh[2][I_CHUNK][HID];   // 2*32*128*4 = 32KB
  __shared__ float w_sh[2][I_CHUNK][JT];     // 1KB
  const int t  = threadIdx.x;                // channel c
  const int j0 = blockIdx.x * JT;
  const unsigned xs_base = (unsigned)(uintptr_t)&xs_sh[0][0][0];
  const unsigned w_base  = (unsigned)(uintptr_t)&w_sh[0][0][0];
  const float inv = 1.0f / total[0];
  const float lec = le[t];
  const int nchunks = n / I_CHUNK;           // 16

  auto issue = [&](int ck) {
    const int buf = ck & 1;
    const int i0  = ck * I_CHUNK;
    // xs chunk: I_CHUNK*HID floats = 1024 float4 -> 8 b128 per thread
#pragma unroll
    for (int q = 0; q < (I_CHUNK * HID / 4) / 128; ++q) {
      const int idx = q * 128 + t;
      const int r = idx >> 5, c4 = idx & 31;   // 32 float4 per row
      async_load_b128(xs_base + (unsigned)(buf * (I_CHUNK * HID * 4) + idx * 16),
                      xs + (size_t)(i0 + r) * HID + (c4 << 2));
    }
    // W chunk: I_CHUNK rows x JT cols = 128 floats -> 1 b32 per thread
    {
      const int r = t >> 2, col = t & 3;
      async_load_b32(w_base + (unsigned)(buf * (I_CHUNK * JT * 4) + t * 4),
                     W + (size_t)(i0 + r) * n + j0 + col);
    }
  };

  issue(0);
  float M[JT], S[JT], T[JT];
#pragma unroll
  for (int jj = 0; jj < JT; ++jj) { M[jj] = -INFINITY; S[jj] = 0.0f; T[jj] = 0.0f; }

  for (int ck = 0; ck < nchunks; ++ck) {
    const int buf = ck & 1;
    if (ck + 1 < nchunks) {
      issue(ck + 1);
      asm volatile("s_wait_asynccnt 0x9" ::: "memory");  // chunk ck drained
    } else {
      asm volatile("s_wait_asynccnt 0x0" ::: "memory");
    }
    __syncthreads();
    for (int i = 0; i < I_CHUNK; ++i) {
      const float xv = xs_sh[buf][i][t];
#pragma unroll
      for (int jj = 0; jj < JT; ++jj) {
        const float wij = w_sh[buf][i][jj] * inv;
        const float m  = fmaxf(xv + wij * lec, 0.0f) + GEN_EPS;
        const float Mn = fmaxf(M[jj], m);
        const float r  = __expf(M[jj] - Mn);   // 0 on first iter (M = -inf)
        const float e  = __expf(m - Mn);
        S[jj] = S[jj] * r + e;
        T[jj] = T[jj] * r + e * m;
        M[jj] = Mn;
      }
    }
    __syncthreads();
  }
#pragma unroll
  for (int jj = 0; jj < JT; ++jj)
    out[(size_t)(j0 + jj) * HID + t] =
        T[jj] / S[jj] + xd[(size_t)(j0 + jj) * HID + t];
}

// ---------- training-mode BatchNorm1d over rows, then ReLU ------------------
__global__ void bn_relu(const float* __restrict__ X, const float* __restrict__ g,
                        const float* __restrict__ b, float* __restrict__ Y,
                        int C, int n) {
  __shared__ float sh[512];
  const int c = blockIdx.x;
  const int t = threadIdx.x;          // blockDim == 256
  float s = 0.0f, q = 0.0f;
  for (int i = t; i < n; i += 256) {
    const float v = X[(size_t)i * C + c];
    s += v; q += v * v;
  }
  sh[t] = s; sh[256 + t] = q; __syncthreads();
  for (int o = 128; o > 0; o >>= 1) {
    if (t < o) { sh[t] += sh[t + o]; sh[256 + t] += sh[256 + t + o]; }
    __syncthreads();
  }
  const float mean = sh[0] / n;
  const float var  = sh[256] / n - mean * mean;
  const float inv  = rsqrtf(var + BN_EPS);
  const float gg = g[c], bb = b[c];
  for (int i = t; i < n; i += 256) {
    const float v = X[(size_t)i * C + c];
    Y[(size_t)i * C + c] = fmaxf((v - mean) * inv * gg + bb, 0.0f);
  }
}

// ---------- attention pooling ----------------------------------------------
__global__ void att_scores(const float* __restrict__ h, const float* __restrict__ attw,
                           const float* __restrict__ attb, float* __restrict__ alpha,
                           int n, int C) {
  __shared__ float sh[512];
  const int i = threadIdx.x;          // blockDim == n == 512
  float s = attb[0];
  for (int c = 0; c < C; ++c) s += h[(size_t)i * C + c] * attw[c];
  sh[i] = s; __syncthreads();
  for (int o = 256; o > 0; o >>= 1) { if (i < o) sh[i] = fmaxf(sh[i], sh[i + o]); __syncthreads(); }
  const float mx = sh[0]; __syncthreads();
  const float e = __expf(s - mx);
  sh[i] = e; __syncthreads();
  for (int o = 256; o > 0; o >>= 1) { if (i < o) sh[i] += sh[i + o]; __syncthreads(); }
  alpha[i] = e / sh[0];
}

__global__ void att_pool(const float* __restrict__ h, const float* __restrict__ alpha,
                         float* __restrict__ out, int n, int C) {
  const int c = threadIdx.x;          // blockDim == C == 128
  float acc = 0.0f;
  for (int i = 0; i < n; ++i) acc += alpha[i] * h[(size_t)i * C + c];
  out[c] = acc;
}

// ---------- misc small kernels ---------------------------------------------
__global__ void rowsum_total(const float* __restrict__ W, float* __restrict__ rowsums,
                             float* __restrict__ total, int n) {
  __shared__ float sh[256];
  const int i = blockIdx.x, t = threadIdx.x;
  float s = 0.0f;
  for (int j = t; j < n; j += 256) s += W[(size_t)i * n + j];
  sh[t] = s; __syncthreads();
  for (int o = 128; o > 0; o >>= 1) { if (t < o) sh[t] += sh[t + o]; __syncthreads(); }
  if (t == 0) { rowsums[i] = sh[0]; atomicAdd(total, sh[0]); }
}

__global__ void build_topo_x(const float* __restrict__ rowsums, const float* __restrict__ total,
                             const float* __restrict__ lsrc, const float* __restrict__ ldst,
                             float* __restrict__ xs, float* __restrict__ xd) {
  const int i = blockIdx.x, c = threadIdx.x;   // blockDim == HID
  const float tx = rowsums[i] / total[0];
  xs[(size_t)i * HID + c] = tx * lsrc[c];
  xd[(size_t)i * HID + c] = tx * ldst[c];
}

__global__ void zero_f32(float* __restrict__ p, int n) {
  const int i = blockIdx.x * blockDim.x + threadIdx.x;
  if (i < n) p[i] = 0.0f;
}

// routing layer 0: y0[k] += sum_{j in chunk} r[j] * w0[j*80+k]
__global__ void rout_layer0(const float* __restrict__ r, const float* __restrict__ w0,
                            float* __restrict__ y0) {
  const int k = threadIdx.x;               // blockDim == 128, active k < 80
  const int jbase = blockIdx.x * 256;
  float acc = 0.0f;
  if (k < 80) {
    for (int j = jbase; j < jbase + 256; ++j)
      acc += r[j] * w0[(size_t)j * 80 + k];
    atomicAdd(&y0[k], acc);
  }
}

__global__ void add_bias_relu(const float* __restrict__ x, const float* __restrict__ b,
                              float* __restrict__ y, int n) {
  const int i = blockIdx.x * blockDim.x + threadIdx.x;
  if (i < n) y[i] = fmaxf(x[i] + b[i], 0.0f);
}

template <bool RELU>
__global__ void dense_vec(const float* __restrict__ x, const float* __restrict__ W,
                          const float* __restrict__ b, float* __restrict__ y,
                          int nin, int nout) {
  const int k = blockIdx.x * blockDim.x + threadIdx.x;
  if (k >= nout) return;
  float acc = b[k];
  for (int j = 0; j < nin; ++j) acc += x[j] * W[(size_t)j * nout + k];
  y[k] = RELU ? fmaxf(acc, 0.0f) : acc;
}

// ---------------------------------------------------------------------------
extern "C" void kernel_launch(void* const* d_in, const int* in_sizes, int n_in,
                              void* d_out, int out_size, void* d_ws, size_t ws_size,
                              hipStream_t stream) {
  (void)in_sizes; (void)n_in; (void)out_size; (void)ws_size;
  const int n = NODES;
  auto in = [&](int i) { return (const float*)d_in[i]; };

  // ---- input leaf indices (setup_inputs insertion order, recursive) ----
  const float* topology = in(0);
  const float* traffic  = in(1);
  const float* routing  = in(2);
  // topo_params (cin=1)
  const float* tp_c0_lsrc = in(3),  *tp_c0_ldst = in(4),  *tp_c0_le = in(5);
  const float* tp_c0_w1 = in(6), *tp_c0_bg = in(7), *tp_c0_bb = in(8), *tp_c0_w2 = in(9);
  const float* tp_c1_le = in(10), *tp_c1_w1 = in(11), *tp_c1_bg = in(12),
             * tp_c1_bb = in(13), *tp_c1_w2 = in(14);
  const float* tp_bn0g = in(15), *tp_bn0b = in(16), *tp_bn1g = in(17), *tp_bn1b = in(18);
  const float* tp_lw[5] = { in(19), in(21), in(23), in(25), in(27) };
  const float* tp_lb[5] = { in(20), in(22), in(24), in(26), in(28) };
  const float* tp_attw = in(29), *tp_attb = in(30);
  // traffic_params (cin=512)
  const float* fp_c0_lsrc = in(31), *fp_c0_ldst = in(32), *fp_c0_le = in(33);
  const float* fp_c0_w1 = in(34), *fp_c0_bg = in(35), *fp_c0_bb = in(36), *fp_c0_w2 = in(37);
  const float* fp_c1_le = in(38), *fp_c1_w1 = in(39), *fp_c1_bg = in(40),
             * fp_c1_bb = in(41), *fp_c1_w2 = in(42);
  const float* fp_bn0g = in(43), *fp_bn0b = in(44), *fp_bn1g = in(45), *fp_bn1b = in(46);
  const float* fp_lw[5] = { in(47), in(49), in(51), in(53), in(55) };
  const float* fp_lb[5] = { in(48), in(50), in(52), in(54), in(56) };
  const float* fp_attw = in(57), *fp_attb = in(58);
  // routing_params
  const float* rw0 = in(59), *rb0 = in(60), *rw1 = in(61), *rb1 = in(62);
  const float* rw2 = in(63), *rb2 = in(64), *rw3 = in(65), *rb3 = in(66);
  // joint_params
  const float* jw0 = in(67), *jb0 = in(68), *jw1 = in(69), *jb1 = in(70);
  const float* jw2 = in(71), *jb2 = in(72);

  // ---- workspace carving ----
  float* Wp = (float*)d_ws;
  size_t off = 0;
  auto alloc = [&](size_t nf) { float* p = Wp + off; off += nf; return p; };
  float* rowsums = alloc(n);
  float* totals  = alloc(4);            // [0]=topo sum, [1]=traffic sum
  float* xsbuf   = alloc((size_t)n * HID);
  float* xdbuf   = alloc((size_t)n * HID);
  float* aggr    = alloc((size_t)n * HID);
  float* mm1     = alloc((size_t)n * 2 * HID);
  float* h1      = alloc((size_t)n * 2 * HID);
  float* hA      = alloc((size_t)n * HID);
  float* hB      = alloc((size_t)n * HID);
  float* alpha   = alloc(n);
  float* cat     = alloc(384);
  float* y0      = alloc(80);
  float* ytA     = alloc(256);
  float* ytB     = alloc(256);

  const int T_MN = (n / 16) * (2 * HID / 16);   // 512x256 -> 512 tiles
  const int T_MH = (n / 16) * (HID / 16);       // 512x128 -> 256 tiles

  // ---- scalars / init ----
  zero_f32<<<1, 128, 0, stream>>>(totals, 4);
  zero_f32<<<1, 128, 0, stream>>>(y0, 80);
  rowsum_total<<<n, 256, 0, stream>>>(topology, rowsums, &totals[0], n);
  build_topo_x<<<n, HID, 0, stream>>>(rowsums, &totals[0], tp_c0_lsrc, tp_c0_ldst,
                                      xsbuf, xdbuf);
  rowsum_total<<<n, 256, 0, stream>>>(traffic, rowsums, &totals[1], n); // rowsums reuse OK (in-order)

  // ---- one GCN branch ----
  auto run_branch = [&](const float* Wmat, const float* totalp,
                        const float* xs0, const float* xd0,
                        const float* c0le, const float* c0w1, const float* c0bg,
                        const float* c0bb, const float* c0w2,
                        const float* c1le, const float* c1w1, const float* c1bg,
                        const float* c1bb, const float* c1w2,
                        const float* bn0g, const float* bn0b,
                        const float* bn1g, const float* bn1b,
                        const float* const* lw, const float* const* lb,
                        const float* attw, const float* attb, float* bout) {
    // conv0
    gen_msg_softmax<<<n / JT, 128, 0, stream>>>(xs0, xd0, Wmat, totalp, c0le, aggr, n);
    gemm_f32_wmma<0><<<T_MN, 32, 0, stream>>>(aggr, c0w1, nullptr, mm1, n, 2 * HID, HID);
    bn_relu<<<2 * HID, 256, 0, stream>>>(mm1, c0bg, c0bb, h1, 2 * HID, n);
    gemm_f32_wmma<0><<<T_MH, 32, 0, stream>>>(h1, c0w2, nullptr, hA, n, HID, 2 * HID);
    bn_relu<<<HID, 256, 0, stream>>>(hA, bn0g, bn0b, hB, HID, n);
    // conv1
    gen_msg_softmax<<<n / JT, 128, 0, stream>>>(hB, hB, Wmat, totalp, c1le, aggr, n);
    gemm_f32_wmma<0><<<T_MN, 32, 0, stream>>>(aggr, c1w1, nullptr, mm1, n, 2 * HID, HID);
    bn_relu<<<2 * HID, 256, 0, stream>>>(mm1, c1bg, c1bb, h1, 2 * HID, n);
    gemm_f32_wmma<0><<<T_MH, 32, 0, stream>>>(h1, c1w2, nullptr, hA, n, HID, 2 * HID);
    bn_relu<<<HID, 256, 0, stream>>>(hA, bn1g, bn1b, hB, HID, n);
    // 5-layer head (relu on first 4)
    float* src = hB; float* dst = hA;
    for (int i = 0; i < 5; ++i) {
      if (i < 4) gemm_f32_wmma<2><<<T_MH, 32, 0, stream>>>(src, lw[i], lb[i], dst, n, HID, HID);
      else       gemm_f32_wmma<1><<<T_MH, 32, 0, stream>>>(src, lw[i], lb[i], dst, n, HID, HID);
      float* t = src; src = dst; dst = t;
    }
    att_scores<<<1, n, 0, stream>>>(src, attw, attb, alpha, n, HID);
    att_pool<<<1, HID, 0, stream>>>(src, alpha, bout, n, HID);
  };

  // topo branch -> cat[0:128]
  run_branch(topology, &totals[0], xsbuf, xdbuf,
             tp_c0_le, tp_c0_w1, tp_c0_bg, tp_c0_bb, tp_c0_w2,
             tp_c1_le, tp_c1_w1, tp_c1_bg, tp_c1_bb, tp_c1_w2,
             tp_bn0g, tp_bn0b, tp_bn1g, tp_bn1b,
             tp_lw, tp_lb, tp_attw, tp_attb, cat + 0);

  // traffic branch: x0 = I  =>  xs = lin_src, xd = lin_dst directly
  run_branch(traffic, &totals[1], fp_c0_lsrc, fp_c0_ldst,
             fp_c0_le, fp_c0_w1, fp_c0_bg, fp_c0_bb, fp_c0_w2,
             fp_c1_le, fp_c1_w1, fp_c1_bg, fp_c1_bb, fp_c1_w2,
             fp_bn0g, fp_bn0b, fp_bn1g, fp_bn1b,
             fp_lw, fp_lb, fp_attw, fp_attb, cat + HID);

  // routing MLP: 262144 -> 80 -> 80 -> 60 -> 128, relu on first 3
  rout_layer0<<<(n * n) / 256, 128, 0, stream>>>(routing, rw0, y0);
  add_bias_relu<<<1, 128, 0, stream>>>(y0, rb0, ytA, 80);
  dense_vec<true ><<<1, 128, 0, stream>>>(ytA, rw1, rb1, ytB, 80, 80);
  dense_vec<true ><<<1, 128, 0, stream>>>(ytB, rw2, rb2, ytA, 80, 60);
  dense_vec<false><<<1, 128, 0, stream>>>(ytA, rw3, rb3, cat + 256, 60, 128);

  // joint MLP: 384 -> 256 -> 128 -> 64 (relu on first 2)
  dense_vec<true ><<<2, 128, 0, stream>>>(cat, jw0, jb0, ytB, 384, 256);
  dense_vec<true ><<<1, 128, 0, stream>>>(ytB, jw1, jb1, ytA, 256, 128);
  dense_vec<false><<<1, 64,  0, stream>>>(ytA, jw2, jb2, (float*)d_out, 128, 64);
}